// attention_75771813036887
// MI455X (gfx1250) — compile-verified
//
#include <hip/hip_runtime.h>
#include <math.h>

// Model dims
#define T_  32
#define B_  64
#define E_  128
#define U_  256
#define D_  512
#define DN_ 1024
#define V_  32000
#define BT_ 2048   // B_*T_
#define G4U 1024   // 4*U_
#define G4D 2048   // 4*D_
#define NWG 32     // decoder persistent workgroups
#define APAD 36    // LDS A-tile row stride (dwords): conflict-free, 16B aligned

typedef float v2f __attribute__((ext_vector_type(2)));
typedef float v8f __attribute__((ext_vector_type(8)));

__device__ __forceinline__ float sigf(float x) { return 1.0f / (1.0f + __expf(-x)); }

__device__ __forceinline__ float wave_sum(float v) {
    #pragma unroll
    for (int o = 16; o > 0; o >>= 1) v += __shfl_xor(v, o, 32);
    return v;
}
__device__ __forceinline__ float wave_max(float v) {
    #pragma unroll
    for (int o = 16; o > 0; o >>= 1) v = fmaxf(v, __shfl_xor(v, o, 32));
    return v;
}

// ---------------------------------------------------------------------------
// f32 GEMM with V_WMMA_F32_16X16X4_F32.   C = act(A[MxK] @ B[KxN] + bias)
// LDB is compile-time: B row offsets become immediate-offset loads and the
// only per-k address math is one pointer increment.
// grid: x = N/128, y = M/64 ; block = 256 (8 waves).
// Each wave owns a 64x16 strip: 4 M-tiles sharing every B fragment (4x reuse).
// Fragment layouts:
//   A: lane L holds A[m = L%16][k = r + 2*(L/16)]   (r = vgpr 0..1)
//   B: lane L holds B[k = r + 2*(L/16)][n = L%16]
//   C: lane L, vgpr r holds C[m = r + 8*(L/16)][n = L%16]
// ---------------------------------------------------------------------------
template <int LDB>
__global__ void gemm_f32_wmma(const float* __restrict__ A, int lda,
                              const float* __restrict__ Bm,
                              const float* __restrict__ bias,
                              float* __restrict__ C, int ldc,
                              int K, int relu)
{
    __shared__ float aT[64][APAD];          // 64 rows x 32 used cols (padded)
    const int tid  = threadIdx.x;
    const int wv   = tid >> 5;
    const int lane = tid & 31;
    const int half = lane >> 4;
    const int m16  = lane & 15;
    const int row0 = blockIdx.y * 64;
    const int col  = blockIdx.x * 128 + wv * 16 + m16;

    // running B pointer: rows (2*half) and (2*half+1), advanced 4 rows / step
    const float* bp = Bm + (size_t)(2 * half) * LDB + col;

    v8f acc[4] = {v8f{}, v8f{}, v8f{}, v8f{}};

    for (int kc = 0; kc < K; kc += 32) {
        __syncthreads();
        // stage 64x32 A tile: 512 float4 / 256 threads = 2 each (b128 in+out)
        #pragma unroll
        for (int it = 0; it < 2; ++it) {
            const int idx = tid + it * 256;          // 0..511
            const int r = idx >> 3, q = idx & 7;     // row, float4 col
            const float4 v = *(const float4*)(A + (size_t)(row0 + r) * lda + kc + q * 4);
            *(float4*)(&aT[r][q * 4]) = v;
        }
        __syncthreads();
        #pragma unroll
        for (int kk = 0; kk < 32; kk += 4) {
            v2f b;
            b.x = bp[0];
            b.y = bp[LDB];            // immediate offset
            bp += 4 * (size_t)LDB;    // one pointer add per k-step
            #pragma unroll
            for (int rb = 0; rb < 4; ++rb) {
                v2f a;
                a.x = aT[rb * 16 + m16][kk + 0 + 2 * half];
                a.y = aT[rb * 16 + m16][kk + 1 + 2 * half];
                acc[rb] = __builtin_amdgcn_wmma_f32_16x16x4_f32(false, a, false, b,
                                                                (short)0, acc[rb],
                                                                false, false);
            }
        }
    }
    const float bv = bias ? bias[col] : 0.0f;
    #pragma unroll
    for (int rb = 0; rb < 4; ++rb) {
        #pragma unroll
        for (int r = 0; r < 8; ++r) {
            float v = acc[rb][r] + bv;
            if (relu) v = fmaxf(v, 0.0f);
            C[(size_t)(row0 + rb * 16 + r + 8 * half) * ldc + col] = v;
        }
    }
}

// ---------------------------------------------------------------------------
// One fused encoder LSTM timestep, both directions.
// grid: x = U_/32 (u-tiles), y = B_/16 (m-tiles), z = 2 (dir); block = 256.
// Each wave: one 16x16 WMMA tile of z = h_prev @ Wh for one gate.
// Gate tiles exchanged through LDS, cell applied in-kernel.
// Xp already contains x@Wx + b.
// ---------------------------------------------------------------------------
__global__ void enc_lstm_step(int s,
    const float* __restrict__ hf_in, float* __restrict__ hf_out, float* __restrict__ cf,
    const float* __restrict__ hb_in, float* __restrict__ hb_out, float* __restrict__ cb,
    const float* __restrict__ XF, const float* __restrict__ XB,
    const float* __restrict__ WhF, const float* __restrict__ WhB,
    float* __restrict__ enc_states)
{
    __shared__ float aT[16][APAD];
    __shared__ float zb[4][16][32];

    const int dir = blockIdx.z;
    const float* h_in = dir ? hb_in : hf_in;
    float*      h_out = dir ? hb_out : hf_out;
    float*      cbuf  = dir ? cb : cf;
    const float* Xp   = dir ? XB : XF;
    const float* Wh   = dir ? WhB : WhF;
    const int t = dir ? (T_ - 1 - s) : s;   // backward scans xs reversed

    const int tid  = threadIdx.x;
    const int wv   = tid >> 5;
    const int lane = tid & 31;
    const int half = lane >> 4;
    const int m16  = lane & 15;
    const int gate = wv & 3;          // i,f,g,o
    const int usub = wv >> 2;         // 0..1
    const int m0   = blockIdx.y * 16;
    const int un   = blockIdx.x * 32 + usub * 16 + m16;  // 0..255 within gate
    const int col  = gate * U_ + un;                     // 0..1023

    const float* bp = Wh + (size_t)(2 * half) * G4U + col;   // running B pointer

    v8f acc = {};
    for (int kc = 0; kc < U_; kc += 32) {
        __syncthreads();
        {   // stage 16x32 h tile (128 float4)
            if (tid < 128) {
                const int r = tid >> 3, q = tid & 7;
                const float4 v = *(const float4*)(h_in + (m0 + r) * U_ + kc + q * 4);
                *(float4*)(&aT[r][q * 4]) = v;
            }
        }
        __syncthreads();
        #pragma unroll
        for (int kk = 0; kk < 32; kk += 4) {
            v2f a, b;
            a.x = aT[m16][kk + 0 + 2 * half];
            a.y = aT[m16][kk + 1 + 2 * half];
            b.x = bp[0];
            b.y = bp[G4U];
            bp += 4 * (size_t)G4U;
            acc = __builtin_amdgcn_wmma_f32_16x16x4_f32(false, a, false, b,
                                                        (short)0, acc, false, false);
        }
    }
    __syncthreads();
    #pragma unroll
    for (int r = 0; r < 8; ++r) {
        const int ml = r + 8 * half;
        float z = acc[r] + Xp[((size_t)(m0 + ml) * T_ + t) * G4U + col];
        zb[gate][ml][usub * 16 + m16] = z;
    }
    __syncthreads();
    #pragma unroll
    for (int e = tid; e < 512; e += 256) {
        const int ml = e >> 5, ul = e & 31;
        const int brow = m0 + ml;
        const int u = blockIdx.x * 32 + ul;
        const float zi = zb[0][ml][ul], zf = zb[1][ml][ul];
        const float zg = zb[2][ml][ul], zo = zb[3][ml][ul];
        const float c_old = cbuf[brow * U_ + u];
        const float cn = sigf(zf) * c_old + sigf(zi) * tanhf(zg);
        const float hn = sigf(zo) * tanhf(cn);
        cbuf[brow * U_ + u]  = cn;
        h_out[brow * U_ + u] = hn;
        enc_states[((size_t)brow * T_ + t) * D_ + dir * U_ + u] = hn;
    }
}

// ---------------------------------------------------------------------------
// ea[b*T+j] = dot(enc_states[b,j,:], Wa[0:512]) + ba   (attention hoist)
// grid = BT_/8, block = 256 (one wave per row).
// ---------------------------------------------------------------------------
__global__ void attn_ea(const float* __restrict__ enc_states,
                        const float* __restrict__ Wa,
                        const float* __restrict__ ba,
                        float* __restrict__ ea)
{
    const int row  = blockIdx.x * 8 + (threadIdx.x >> 5);
    const int lane = threadIdx.x & 31;
    const float* ep = enc_states + (size_t)row * D_;
    float s = 0.f;
    #pragma unroll 4
    for (int k = lane; k < D_; k += 32) s += ep[k] * Wa[k];
    s = wave_sum(s);
    if (lane == 0) ea[row] = s + ba[0];
}

// ---------------------------------------------------------------------------
// Monotonic-epoch grid barrier (NWG co-resident WGs, agent scope).
// Counter is zeroed by hipMemsetAsync before launch -> graph-replay safe.
// ---------------------------------------------------------------------------
__device__ __forceinline__ void grid_barrier(unsigned* cnt, unsigned nwg, unsigned& epoch)
{
    __syncthreads();
    epoch++;
    if (threadIdx.x == 0) {
        __hip_atomic_fetch_add(cnt, 1u, __ATOMIC_RELEASE, __HIP_MEMORY_SCOPE_AGENT);
        while (__hip_atomic_load(cnt, __ATOMIC_ACQUIRE, __HIP_MEMORY_SCOPE_AGENT)
               < nwg * epoch) {
            __builtin_amdgcn_s_sleep(8);
        }
    }
    __syncthreads();
}

// ---------------------------------------------------------------------------
// Persistent decoder: 2048 strictly sequential attention+LSTM steps.
// NWG workgroups x 256 threads. WG w owns gate columns [w*64, w*64+64)
// and hidden cells [w*16, w*16+16). Two grid barriers per step.
// DX already contains x_emb @ dec_Wx[0:128] + dec_b.
// ---------------------------------------------------------------------------
__global__ void decoder_persistent(
    const float* __restrict__ DX,          // BT_ x 2048
    const float* __restrict__ Wc,          // dec_Wx rows 128..639  (512 x 2048)
    const float* __restrict__ Wh,          // dec_Wh                (512 x 2048)
    const float* __restrict__ ea,          // BT_
    const float* __restrict__ enc_states,  // BT_ x 512
    const float* __restrict__ Wa2,         // Wa + 512 (h part)
    const float* __restrict__ Ws, const float* __restrict__ bs,
    const float* __restrict__ hf_fin, const float* __restrict__ cf_fin,
    const float* __restrict__ hb_fin, const float* __restrict__ cb_fin,
    float* __restrict__ gh,                // 512 (global h carry)
    float* __restrict__ gates,             // 2048
    float* __restrict__ hs,                // BT_ x 512
    unsigned* __restrict__ bar)
{
    __shared__ float h_l[D_];
    __shared__ float ctx[D_];
    __shared__ float wj[T_];
    __shared__ float part[4][64];
    __shared__ float red[8];
    __shared__ float hterm_s;
    __shared__ float c_loc[16];

    const int w    = blockIdx.x;
    const int tid  = threadIdx.x;
    const int lane = tid & 31;
    const int wv   = tid >> 5;
    unsigned epoch = 0;

    // init carry: h_t = concat(hf[63], hb[63]); c_t likewise
    if (tid < 16) {
        const int u = w * 16 + tid;
        gh[u]      = (u < U_) ? hf_fin[63 * U_ + u] : hb_fin[63 * U_ + (u - U_)];
        c_loc[tid] = (u < U_) ? cf_fin[63 * U_ + u] : cb_fin[63 * U_ + (u - U_)];
    }
    grid_barrier(bar, NWG, epoch);

    const float Ws0 = Ws[0], bs0 = bs[0];

    for (int step = 0; step < BT_; ++step) {
        // ---- (A) load h, attention scores, softmax, context ----
        h_l[tid]       = gh[tid];
        h_l[tid + 256] = gh[tid + 256];
        __syncthreads();

        float p = h_l[tid] * Wa2[tid] + h_l[tid + 256] * Wa2[tid + 256];
        p = wave_sum(p);
        if (lane == 0) red[wv] = p;
        __syncthreads();
        if (tid == 0) {
            float s = 0.f;
            #pragma unroll
            for (int i = 0; i < 8; ++i) s += red[i];
            hterm_s = s;
        }
        __syncthreads();

        const int bi = step >> 5;          // enc_idx = repeat(arange(B), T)
        if (wv == 0) {
            const float a  = tanhf(ea[bi * T_ + lane] + hterm_s);
            const float sc = a * Ws0 + bs0;
            const float mx = wave_max(sc);
            const float ex = __expf(sc - mx);
            const float sm = wave_sum(ex);
            wj[lane] = ex / sm;
        }
        __syncthreads();

        {   // ctx[k] = sum_j wj[j] * enc_states[bi, j, k]
            float c0 = 0.f, c1 = 0.f;
            const float* eb = enc_states + (size_t)bi * T_ * D_;
            #pragma unroll 4
            for (int j = 0; j < T_; ++j) {
                const float wjj = wj[j];
                c0 += wjj * eb[j * D_ + tid];
                c1 += wjj * eb[j * D_ + tid + 256];
            }
            ctx[tid]       = c0;
            ctx[tid + 256] = c1;
        }
        __syncthreads();

        // ---- (B) gates: z = DX[step] + ctx@Wc + h@Wh (our 64-column slice) ----
        {
            const int colb = w * 64;
            const int cl   = tid & 63;
            const int kp   = tid >> 6;        // 4 K-partials of 256
            const int col  = colb + cl;
            const int k0   = kp * 256;
            float acc = 0.f;
            if (kp < 2) {
                const float* wp = Wc + (size_t)k0 * G4D + col;
                const float* kv = ctx + k0;
                for (int k = 0; k < 256; ++k) acc += kv[k] * wp[(size_t)k * G4D];
            } else {
                const float* wp = Wh + (size_t)(k0 - 512) * G4D + col;
                const float* kv = h_l + (k0 - 512);
                for (int k = 0; k < 256; ++k) acc += kv[k] * wp[(size_t)k * G4D];
            }
            part[kp][cl] = acc;
            __syncthreads();
            if (tid < 64) {
                const float g = part[0][tid] + part[1][tid] + part[2][tid] + part[3][tid]
                              + DX[(size_t)step * G4D + colb + tid];
                gates[colb + tid] = g;
            }
        }
        grid_barrier(bar, NWG, epoch);

        // ---- (C) cell for our 16 hidden units ----
        if (tid < 16) {
            const int u  = w * 16 + tid;
            const float zi = gates[u],        zf = gates[u + 512];
            const float zg = gates[u + 1024], zo = gates[u + 1536];
            const float cn = sigf(zf) * c_loc[tid] + sigf(zi) * tanhf(zg);
            const float hn = sigf(zo) * tanhf(cn);
            c_loc[tid] = cn;
            gh[u] = hn;
            hs[(size_t)step * D_ + u] = hn;
        }
        grid_barrier(bar, NWG, epoch);
    }
}

// ---------------------------------------------------------------------------
extern "C" void kernel_launch(void* const* d_in, const int* in_sizes, int n_in,
                              void* d_out, int out_size, void* d_ws, size_t ws_size,
                              hipStream_t stream)
{
    const float* enc_inp  = (const float*)d_in[0];
    const float* dec_inp  = (const float*)d_in[1];
    const float* enc_Wx_f = (const float*)d_in[2];
    const float* enc_Wh_f = (const float*)d_in[3];
    const float* enc_b_f  = (const float*)d_in[4];
    const float* enc_Wx_b = (const float*)d_in[5];
    const float* enc_Wh_b = (const float*)d_in[6];
    const float* enc_b_b  = (const float*)d_in[7];
    const float* dec_Wx   = (const float*)d_in[8];
    const float* dec_Wh   = (const float*)d_in[9];
    const float* dec_b    = (const float*)d_in[10];
    const float* Wa       = (const float*)d_in[11];
    const float* ba       = (const float*)d_in[12];
    const float* Ws       = (const float*)d_in[13];
    const float* bs       = (const float*)d_in[14];
    const float* Wd       = (const float*)d_in[15];
    const float* bd       = (const float*)d_in[16];
    const float* Wo       = (const float*)d_in[17];
    const float* bo       = (const float*)d_in[18];
    float* out = (float*)d_out;

    // ---- workspace layout (floats), ~38 MB total ----
    float* ws  = (float*)d_ws;
    float* XF  = ws;                                  // 2048*1024
    float* XB  = XF + (size_t)BT_ * G4U;              // 2048*1024
    float* DX  = XB + (size_t)BT_ * G4U;              // 2048*2048
    float* ES  = DX + (size_t)BT_ * G4D;              // enc_states 2048*512
    float* hf0 = ES + (size_t)BT_ * D_;               // 64*256 each:
    float* hf1 = hf0 + B_ * U_;
    float* hb0 = hf1 + B_ * U_;
    float* hb1 = hb0 + B_ * U_;
    float* cf  = hb1 + B_ * U_;
    float* cb  = cf  + B_ * U_;
    float* ea  = cb  + B_ * U_;                       // 2048
    float* gh  = ea  + BT_;                           // 512
    float* gts = gh  + D_;                            // 2048
    unsigned* bar = (unsigned*)(gts + G4D);           // 64B barrier counter
    float* hs     = XB;   // reuse: XB dead after encoder, hs needs 2048*512
    float* hidden = XF;   // reuse: XF dead after encoder, hidden needs 2048*1024

    // zero h/c state (6 contiguous 64x256 buffers) and barrier counter
    hipMemsetAsync(hf0, 0, (size_t)6 * B_ * U_ * sizeof(float), stream);
    hipMemsetAsync(bar, 0, 64, stream);

    // ---- stage A: hoisted projections (f32 WMMA GEMMs) ----
    // XF = enc_inp @ enc_Wx_f + b_f ;  XB = enc_inp @ enc_Wx_b + b_b
    gemm_f32_wmma<G4U><<<dim3(G4U / 128, BT_ / 64), 256, 0, stream>>>(
        enc_inp, E_, enc_Wx_f, enc_b_f, XF, G4U, E_, 0);
    gemm_f32_wmma<G4U><<<dim3(G4U / 128, BT_ / 64), 256, 0, stream>>>(
        enc_inp, E_, enc_Wx_b, enc_b_b, XB, G4U, E_, 0);
    // DX = dec_inp @ dec_Wx[0:128,:] + dec_b
    gemm_f32_wmma<G4D><<<dim3(G4D / 128, BT_ / 64), 256, 0, stream>>>(
        dec_inp, E_, dec_Wx, dec_b, DX, G4D, E_, 0);

    // ---- stage B: bidirectional encoder recurrence (h ping-pongs) ----
    for (int s = 0; s < T_; ++s) {
        const float* hfi = (s & 1) ? hf1 : hf0;  float* hfo = (s & 1) ? hf0 : hf1;
        const float* hbi = (s & 1) ? hb1 : hb0;  float* hbo = (s & 1) ? hb0 : hb1;
        enc_lstm_step<<<dim3(U_ / 32, B_ / 16, 2), 256, 0, stream>>>(
            s, hfi, hfo, cf, hbi, hbo, cb, XF, XB, enc_Wh_f, enc_Wh_b, ES);
    }
    // T_=32 even -> final h in buffer 0
    const float* hf_fin = hf0;
    const float* hb_fin = hb0;

    // ---- stage C: attention hoist + persistent decoder ----
    attn_ea<<<BT_ / 8, 256, 0, stream>>>(ES, Wa, ba, ea);
    decoder_persistent<<<NWG, 256, 0, stream>>>(
        DX, dec_Wx + (size_t)E_ * G4D, dec_Wh, ea, ES, Wa + D_, Ws, bs,
        hf_fin, cf, hb_fin, cb, gh, gts, hs, bar);

    // ---- stage D: output MLP (dominant GEMMs) ----
    // hidden = relu(hs @ Wd + bd)
    gemm_f32_wmma<DN_><<<dim3(DN_ / 128, BT_ / 64), 256, 0, stream>>>(
        hs, D_, Wd, bd, hidden, DN_, D_, 1);
    // logits = hidden @ Wo + bo  -> d_out (2048 x 32000 f32)
    gemm_f32_wmma<V_><<<dim3(V_ / 128, BT_ / 64), 256, 0, stream>>>(
        hidden, DN_, Wo, bo, out, V_, DN_, 0);

    (void)in_sizes; (void)n_in; (void)out_size; (void)ws_size;
}